// CGClass_19310172962924
// MI455X (gfx1250) — compile-verified
//
#include <hip/hip_runtime.h>
#include <math.h>

// ---------------------------------------------------------------------------
// CG-SENSE (MRI) solve on MI455X / gfx1250.
// FFT2 realized as dense ortho-DFT GEMMs on the fp32 WMMA pipe:
//   FFT2(X) = W * X * W,  IFFT2(K) = conj(W) * K * conj(W),  W symmetric.
// Complex GEMM: V_WMMA_F32_16X16X4_F32, 4 waves/block, planar (split re/im)
// operands so A fragments load directly into WMMA pair layout, and an
// LDS-staged deinterleaved + pre-negated B panel -> hot loop is loads+WMMA
// only (no packing movs, no xors).
// ---------------------------------------------------------------------------

#define NFFT     384
#define HW       (NFFT * NFFT)          // 147456
#define NCOIL    16
#define NBATCH   8
#define MAX_IT   10
#define CG_TOL   1e-10f
#define NB_RED   256                    // fixed partial count (deterministic)
#define RED_T    256
#define BPANEL   ((NFFT / 2) * 16)      // 3072 pairs per LDS plane

typedef __attribute__((ext_vector_type(2))) float v2f;
typedef __attribute__((ext_vector_type(8))) float v8f;

// ---- DFT matrix generation (planar): W[n,k] = exp(-2*pi*i*n*k/N)/sqrt(N) --
__global__ void k_gen_w(float* __restrict__ WfRe, float* __restrict__ WfIm,
                        float* __restrict__ WiRe, float* __restrict__ WiIm) {
    int idx = blockIdx.x * blockDim.x + threadIdx.x;
    if (idx >= HW) return;
    int n = idx / NFFT, k = idx % NFFT;
    int m = (n * k) % NFFT;                     // phase is periodic in N
    const float s = 0.05103103630798288f;       // 1/sqrt(384)
    float ang = -2.0f * 3.14159265358979323846f * (float)m / (float)NFFT;
    float c = cosf(ang) * s, sn = sinf(ang) * s;
    WfRe[idx] = c;  WfIm[idx] = sn;             // forward
    WiRe[idx] = c;  WiIm[idx] = -sn;            // inverse = conj
}

// ---- Complex GEMM C = A*B (384x384 complex, planar), batch = blockIdx.z ---
// Block = 128 threads = 4 waves; block tile = 64(M) x 16(N).
// WMMA wave32 layouts (ISA 7.12.2):
//   A 16x4 f32 : lanes 0-15 rows with K{0,1}; lanes 16-31 rows with K{2,3}
//   B 4x16 f32 : lanes 0-15 cols with K{0,1}; lanes 16-31 cols with K{2,3}
//   C/D 16x16  : VGPR v -> row v (lanes 0-15) / v+8 (lanes 16-31), col=lane&15
// Planar A rows make {re(k0),re(k0+1)} a single contiguous b64 load; the B
// panel is staged once per block into LDS as {.,.} pairs (re / im / -im).
__global__ void __launch_bounds__(128) k_cgemm(
        float* __restrict__ Cre, float* __restrict__ Cim,
        const float* __restrict__ Are, const float* __restrict__ Aim,
        const float* __restrict__ Bre, const float* __restrict__ Bim,
        unsigned aStride, unsigned bStride, unsigned cStride) {
    __shared__ float2 sBr[BPANEL];   // {re(2kp,n),  re(2kp+1,n)}
    __shared__ float2 sBi[BPANEL];   // {im(2kp,n),  im(2kp+1,n)}
    __shared__ float2 sBn[BPANEL];   // {-im(2kp,n), -im(2kp+1,n)}

    Are += (size_t)blockIdx.z * aStride;  Aim += (size_t)blockIdx.z * aStride;
    Bre += (size_t)blockIdx.z * bStride;  Bim += (size_t)blockIdx.z * bStride;
    Cre += (size_t)blockIdx.z * cStride;  Cim += (size_t)blockIdx.z * cStride;

    const int tid = threadIdx.x;
    const int n0  = blockIdx.x * 16;

    // Stage + deinterleave the B panel (uniform trip count: 3072/128 = 24).
    for (int pidx = tid; pidx < BPANEL; pidx += 128) {
        const int kp = pidx >> 4;          // K pair index
        const int n  = pidx & 15;
        const size_t g = (size_t)(2 * kp) * NFFT + n0 + n;
        const float r0 = Bre[g], r1 = Bre[g + NFFT];
        const float i0 = Bim[g], i1 = Bim[g + NFFT];
        sBr[pidx] = make_float2(r0, r1);
        sBi[pidx] = make_float2(i0, i1);
        sBn[pidx] = make_float2(-i0, -i1);
    }
    __syncthreads();

    const int wave = tid >> 5;
    const int lane = tid & 31;
    const int half = lane >> 4;
    const int l16  = lane & 15;
    const int m    = blockIdx.y * 64 + wave * 16 + l16;   // A row of this lane

    v8f accRe = {0.f, 0.f, 0.f, 0.f, 0.f, 0.f, 0.f, 0.f};
    v8f accIm = {0.f, 0.f, 0.f, 0.f, 0.f, 0.f, 0.f, 0.f};

    for (int kk = 0; kk < NFFT; kk += 4) {
        const int k0 = kk + 2 * half;
        // A fragments: contiguous planar pairs, already in WMMA layout
        const v2f ar = *(const v2f*)(Are + (size_t)m * NFFT + k0);
        const v2f ai = *(const v2f*)(Aim + (size_t)m * NFFT + k0);
        // B fragments straight from LDS, already in WMMA pair layout
        const int bo = (k0 >> 1) * 16 + l16;
        const v2f br = *(const v2f*)&sBr[bo];
        const v2f bi = *(const v2f*)&sBi[bo];
        const v2f bn = *(const v2f*)&sBn[bo];

        // Re += Ar*Br + Ai*(-Bi) ; Im += Ar*Bi + Ai*Br
        accRe = __builtin_amdgcn_wmma_f32_16x16x4_f32(false, ar, false, br,
                                                      (short)0, accRe, false, false);
        accRe = __builtin_amdgcn_wmma_f32_16x16x4_f32(false, ai, false, bn,
                                                      (short)0, accRe, false, false);
        accIm = __builtin_amdgcn_wmma_f32_16x16x4_f32(false, ar, false, bi,
                                                      (short)0, accIm, false, false);
        accIm = __builtin_amdgcn_wmma_f32_16x16x4_f32(false, ai, false, br,
                                                      (short)0, accIm, false, false);
    }

#pragma unroll
    for (int v = 0; v < 8; ++v) {
        const int row = blockIdx.y * 64 + wave * 16 + v + 8 * half;
        Cre[(size_t)row * NFFT + n0 + l16] = accRe[v];
        Cim[(size_t)row * NFFT + n0 + l16] = accIm[v];
    }
}

// ---- elementwise kernels (grid.y = coil; T planes are planar) -------------
__global__ void k_smul(float* __restrict__ Tre, float* __restrict__ Tim,
                       const float2* __restrict__ s, const float2* __restrict__ p) {
    int i = blockIdx.x * blockDim.x + threadIdx.x;
    size_t off = (size_t)blockIdx.y * HW + i;
    float2 sv = s[off], pv = p[i];
    Tre[off] = sv.x * pv.x - sv.y * pv.y;
    Tim[off] = sv.x * pv.y + sv.y * pv.x;
}

__global__ void k_masky(float* __restrict__ Tre, float* __restrict__ Tim,
                        const float2* __restrict__ y, const int* __restrict__ mask) {
    int i = blockIdx.x * blockDim.x + threadIdx.x;
    size_t off = (size_t)blockIdx.y * HW + i;
    float mf = (float)mask[i];
    float2 v = y[off];
    Tre[off] = v.x * mf;
    Tim[off] = v.y * mf;
}

__global__ void k_maskip(float* __restrict__ Tre, float* __restrict__ Tim,
                         const int* __restrict__ mask) {
    int i = blockIdx.x * blockDim.x + threadIdx.x;
    size_t off = (size_t)blockIdx.y * HW + i;
    float mf = (float)mask[i];
    Tre[off] *= mf;
    Tim[off] *= mf;
}

// out = sum_c conj(s[c])*T[c] + lam*other      (coil combine + Tikhonov)
__global__ void k_ahsum(float2* __restrict__ out,
                        const float* __restrict__ Tre, const float* __restrict__ Tim,
                        const float2* __restrict__ s, const float2* __restrict__ other,
                        const float* __restrict__ lam) {
    int i = blockIdx.x * blockDim.x + threadIdx.x;
    float accx = 0.f, accy = 0.f;
#pragma unroll
    for (int c = 0; c < NCOIL; ++c) {
        size_t off = (size_t)c * HW + i;
        float2 sv = s[off];
        float tre = Tre[off], tim = Tim[off];
        accx += sv.x * tre + sv.y * tim;    // Re(conj(s)*t)
        accy += sv.x * tim - sv.y * tre;    // Im(conj(s)*t)
    }
    const float l = lam[0];
    float2 o = other[i];
    out[i] = make_float2(accx + l * o.x, accy + l * o.y);
}

__global__ void k_initcg(float2* __restrict__ p, float2* __restrict__ x,
                         const float2* __restrict__ r) {
    int i = blockIdx.x * blockDim.x + threadIdx.x;
    p[i] = r[i];
    x[i] = make_float2(0.f, 0.f);
}

// ---- deterministic two-stage reductions -----------------------------------
__global__ void k_dotp(const float2* __restrict__ a, const float2* __restrict__ b,
                       float* __restrict__ partials) {
    __shared__ float sre[RED_T], sim[RED_T];
    float re = 0.f, im = 0.f;
    for (int i = blockIdx.x * RED_T + threadIdx.x; i < HW; i += NB_RED * RED_T) {
        float2 av = a[i], bv = b[i];
        re += av.x * bv.x + av.y * bv.y;     // conj(a)*b
        im += av.x * bv.y - av.y * bv.x;
    }
    int t = threadIdx.x;
    sre[t] = re; sim[t] = im; __syncthreads();
    for (int s = RED_T / 2; s > 0; s >>= 1) {
        if (t < s) { sre[t] += sre[t + s]; sim[t] += sim[t + s]; }
        __syncthreads();
    }
    if (t == 0) { partials[2 * blockIdx.x] = sre[0]; partials[2 * blockIdx.x + 1] = sim[0]; }
}

// scal: [0]=rTr [1]=alpha_re [2]=alpha_im [3]=beta [4]=active
// mode 0: rTr0 = re ; mode 1: alpha = rTr/<p,q>, active = rTr>tol ;
// mode 2: beta = rTr_n/rTr, rTr = rTr_n if active (freeze semantics)
__global__ void k_finalize(const float* __restrict__ partials,
                           float* __restrict__ scal, int mode) {
    __shared__ float sre[NB_RED], sim[NB_RED];
    int t = threadIdx.x;
    sre[t] = partials[2 * t]; sim[t] = partials[2 * t + 1];
    __syncthreads();
    for (int s = NB_RED / 2; s > 0; s >>= 1) {
        if (t < s) { sre[t] += sre[t + s]; sim[t] += sim[t + s]; }
        __syncthreads();
    }
    if (t == 0) {
        float re = sre[0], im = sim[0];
        if (mode == 0) {
            scal[0] = re;
        } else if (mode == 1) {
            float rtr = scal[0];
            float den = re * re + im * im;
            scal[1] =  rtr * re / den;       // alpha = rTr * conj(pq) / |pq|^2
            scal[2] = -rtr * im / den;
            scal[4] = (rtr > CG_TOL) ? 1.f : 0.f;
        } else {
            float rtr_n = re;
            scal[3] = rtr_n / scal[0];
            if (scal[4] != 0.f) scal[0] = rtr_n;
        }
    }
}

// x += alpha*p ; r -= alpha*q (if active), and fused |r|^2 partials
__global__ void k_update_xr(float2* __restrict__ x, float2* __restrict__ r,
                            const float2* __restrict__ p, const float2* __restrict__ q,
                            const float* __restrict__ scal, float* __restrict__ partials) {
    __shared__ float sr[RED_T];
    const float are = scal[1], aim = scal[2];
    const bool act = scal[4] != 0.f;
    float acc = 0.f;
    for (int i = blockIdx.x * RED_T + threadIdx.x; i < HW; i += NB_RED * RED_T) {
        float2 rv = r[i];
        if (act) {
            float2 pv = p[i], qv = q[i], xv = x[i];
            xv.x += are * pv.x - aim * pv.y;
            xv.y += are * pv.y + aim * pv.x;
            rv.x -= are * qv.x - aim * qv.y;
            rv.y -= are * qv.y + aim * qv.x;
            x[i] = xv; r[i] = rv;
        }
        acc += rv.x * rv.x + rv.y * rv.y;
    }
    int t = threadIdx.x;
    sr[t] = acc; __syncthreads();
    for (int s = RED_T / 2; s > 0; s >>= 1) {
        if (t < s) sr[t] += sr[t + s];
        __syncthreads();
    }
    if (t == 0) { partials[2 * blockIdx.x] = sr[0]; partials[2 * blockIdx.x + 1] = 0.f; }
}

__global__ void k_update_p(float2* __restrict__ p, const float2* __restrict__ r,
                           const float* __restrict__ scal) {
    int i = blockIdx.x * blockDim.x + threadIdx.x;
    if (scal[4] != 0.f) {
        float beta = scal[3];
        float2 rv = r[i], pv = p[i];
        p[i] = make_float2(rv.x + beta * pv.x, rv.y + beta * pv.y);
    }
}

__global__ void k_copyout(float2* __restrict__ out, const float2* __restrict__ x) {
    int i = blockIdx.x * blockDim.x + threadIdx.x;
    out[i] = x[i];
}

// ---------------------------------------------------------------------------
extern "C" void kernel_launch(void* const* d_in, const int* in_sizes, int n_in,
                              void* d_out, int out_size, void* d_ws, size_t ws_size,
                              hipStream_t stream) {
    const float* lam  = (const float*)d_in[0];      // [1]
    const float* x_in = (const float*)d_in[1];      // [B,H,W,2]
    const float* y_in = (const float*)d_in[2];      // [B,C,H,W,2]
    const float* s_in = (const float*)d_in[3];      // [B,C,H,W,2]
    const int*   mask = (const int*)d_in[4];        // [H,W]
    (void)d_in[5]; (void)in_sizes; (void)n_in; (void)out_size; (void)ws_size;

    float* ws = (float*)d_ws;
    size_t o = 0;
    float* WfRe = ws + o; o += (size_t)HW;
    float* WfIm = ws + o; o += (size_t)HW;
    float* WiRe = ws + o; o += (size_t)HW;
    float* WiIm = ws + o; o += (size_t)HW;
    float* T0re = ws + o; o += (size_t)NCOIL * HW;
    float* T0im = ws + o; o += (size_t)NCOIL * HW;
    float* T1re = ws + o; o += (size_t)NCOIL * HW;
    float* T1im = ws + o; o += (size_t)NCOIL * HW;
    float2* r = (float2*)(ws + o); o += (size_t)HW * 2;
    float2* p = (float2*)(ws + o); o += (size_t)HW * 2;
    float2* x = (float2*)(ws + o); o += (size_t)HW * 2;
    float2* q = (float2*)(ws + o); o += (size_t)HW * 2;
    float* partials = ws + o; o += 2 * NB_RED;
    float* scal     = ws + o; o += 8;

    const dim3 gE(HW / 256, NCOIL);                 // coil elementwise
    const dim3 gP(HW / 256);                        // pixel elementwise
    const dim3 gG(NFFT / 16, NFFT / 64, NCOIL);     // GEMM: 16-col x 64-row tiles

    k_gen_w<<<HW / 256, 256, 0, stream>>>(WfRe, WfIm, WiRe, WiIm);

    for (int b = 0; b < NBATCH; ++b) {
        const float2* yb = (const float2*)(y_in + (size_t)b * NCOIL * HW * 2);
        const float2* sb = (const float2*)(s_in + (size_t)b * NCOIL * HW * 2);
        const float2* xb = (const float2*)(x_in + (size_t)b * HW * 2);

        // rhs = AH(y) + lam*x0 :  T0 = mask*y ; T0 = Wi*(T0*Wi) ; combine
        k_masky<<<gE, 256, 0, stream>>>(T0re, T0im, yb, mask);
        k_cgemm<<<gG, 128, 0, stream>>>(T1re, T1im, T0re, T0im, WiRe, WiIm, HW, 0, HW);
        k_cgemm<<<gG, 128, 0, stream>>>(T0re, T0im, WiRe, WiIm, T1re, T1im, 0, HW, HW);
        k_ahsum<<<gP, 256, 0, stream>>>(r, T0re, T0im, sb, xb, lam);
        k_initcg<<<gP, 256, 0, stream>>>(p, x, r);
        k_dotp<<<NB_RED, RED_T, 0, stream>>>(r, r, partials);
        k_finalize<<<1, NB_RED, 0, stream>>>(partials, scal, 0);

        for (int it = 0; it < MAX_IT; ++it) {
            // q = AH(A(p)) + lam*p
            k_smul<<<gE, 256, 0, stream>>>(T0re, T0im, sb, p);
            k_cgemm<<<gG, 128, 0, stream>>>(T1re, T1im, T0re, T0im, WfRe, WfIm, HW, 0, HW);
            k_cgemm<<<gG, 128, 0, stream>>>(T0re, T0im, WfRe, WfIm, T1re, T1im, 0, HW, HW);
            k_maskip<<<gE, 256, 0, stream>>>(T0re, T0im, mask);   // mask^2 == mask (0/1)
            k_cgemm<<<gG, 128, 0, stream>>>(T1re, T1im, T0re, T0im, WiRe, WiIm, HW, 0, HW);
            k_cgemm<<<gG, 128, 0, stream>>>(T0re, T0im, WiRe, WiIm, T1re, T1im, 0, HW, HW);
            k_ahsum<<<gP, 256, 0, stream>>>(q, T0re, T0im, sb, p, lam);

            k_dotp<<<NB_RED, RED_T, 0, stream>>>(p, q, partials);
            k_finalize<<<1, NB_RED, 0, stream>>>(partials, scal, 1);
            k_update_xr<<<NB_RED, RED_T, 0, stream>>>(x, r, p, q, scal, partials);
            k_finalize<<<1, NB_RED, 0, stream>>>(partials, scal, 2);
            k_update_p<<<gP, 256, 0, stream>>>(p, r, scal);
        }
        k_copyout<<<gP, 256, 0, stream>>>((float2*)d_out + (size_t)b * HW, x);
    }
}